// GroupwiseCostVolume3D_76338748719680
// MI455X (gfx1250) — compile-verified
//
#include <hip/hip_runtime.h>

// Groupwise cost volume on MI455X (gfx1250, wave32).
// cost[n,g,d,h,w] = mean_c( L[n,g*8+c,h,w] * R[n,g*8+c,h,w-d] ), 1.0 where w<d.
//
// Memory-bound (~420 MB traffic, ~1.26 GFLOP, AI~3) -> HBM floor ~18us at
// 23.3 TB/s. Instruction-issue on the D-fragment scatter is nearly
// co-limiting, so this version makes that path branchless:
//  - 16x16 GEMM tiles via V_WMMA_F32_16X16X4_F32 (exact f32; K=4 x2 = cg 8)
//  - band correlation = 4 shifted GEMMs per 16-wide w tile (off = 0,16,32,48)
//  - D->LDS scatter uses a v_min_u32 "dump row" clamp (no exec branching)
//  - 1/8 mean folded into the B fragment (4 muls/wave, not 32 muls/lane)
//  - w<d masking + OOB column guard only exist in the EDGE=true instantiation
//    (w0 < 48, i.e. 3 of 20 w-tiles)
//  - writeout streams LDS -> HBM as coalesced b128 (float4) rows.

typedef __attribute__((ext_vector_type(2))) float v2f;
typedef __attribute__((ext_vector_type(8))) float v8f;

constexpr int N_   = 4;
constexpr int C_   = 64;
constexpr int H_   = 160;
constexpr int W_   = 320;
constexpr int G_   = 8;
constexpr int CG_  = 8;
constexpr int DMAX = 48;
constexpr int HW   = H_ * W_;
constexpr int TILE = 16;
constexpr int DROWS = DMAX + 1;          // row 48 = garbage dump for invalid d

template <bool EDGE>
__global__ __launch_bounds__(256)
void cost_volume_wmma(const float* __restrict__ l,
                      const float* __restrict__ r,
                      float* __restrict__ out,
                      int wtile0) {
  __shared__ __align__(16) float tile_out[G_][DROWS][TILE];   // 24.5 KB

  const int lane = threadIdx.x & 31;
  const int g    = threadIdx.x >> 5;           // one group per wave (8 waves)
  const int w0   = (wtile0 + blockIdx.x) * TILE;
  const int h    = blockIdx.y;
  const int n    = blockIdx.z;

  const int jcol  = lane & 15;                 // N (output w) / A row index M
  const int khalf = (lane >> 4) * 2;           // K: lanes 0-15 -> {0,1}, 16-31 -> {2,3}

  const float* lb = l + (((size_t)n * C_ + g * CG_) * H_ + h) * W_;
  const float* rb = r + (((size_t)n * C_ + g * CG_) * H_ + h) * W_;

  // B fragments (left features), pre-scaled by 1/8 so D is already the mean.
  v2f b0, b1;
  b0.x = (1.0f / CG_) * lb[(0 + khalf) * HW + w0 + jcol];
  b0.y = (1.0f / CG_) * lb[(1 + khalf) * HW + w0 + jcol];
  b1.x = (1.0f / CG_) * lb[(4 + khalf) * HW + w0 + jcol];
  b1.y = (1.0f / CG_) * lb[(5 + khalf) * HW + w0 + jcol];

  float* const ldsg = &tile_out[g][0][0];

#pragma unroll
  for (int off = 0; off < 64; off += 16) {
    // A fragment (right features): A[i][k] = R[c0+k, w0-off+i].
    // A column (w0-off+i) equals (w - d) for every output it feeds, so OOB
    // columns only touch entries the reference masks to 1.0 -> load 0.
    const int col = w0 - off + jcol;
    v2f a0 = {0.0f, 0.0f};
    v2f a1 = {0.0f, 0.0f};
    if (!EDGE || col >= 0) {
      a0.x = rb[(0 + khalf) * HW + col];
      a0.y = rb[(1 + khalf) * HW + col];
      a1.x = rb[(4 + khalf) * HW + col];
      a1.y = rb[(5 + khalf) * HW + col];
    }

    v8f c = {};
    // 8 args: (neg_a, A, neg_b, B, c_mod, C, reuse_a, reuse_b)
    c = __builtin_amdgcn_wmma_f32_16x16x4_f32(false, a0, false, b0, (short)0, c,
                                              false, false);
    c = __builtin_amdgcn_wmma_f32_16x16x4_f32(false, a1, false, b1, (short)0, c,
                                              false, false);

    // D layout: VGPR v, lane -> M = v + 8*(lane>>4), N = lane&15.
    // d = off + jcol - M decreases by 1 per v; invalid d clamps to dump row.
    const int dbase = off + jcol - ((lane >> 4) * 8);
#pragma unroll
    for (int v = 0; v < 8; ++v) {
      const int d = dbase - v;
      const unsigned row = min((unsigned)d, (unsigned)DMAX);  // v_min_u32 clamp
      float val = c[v];
      if (EDGE) val = (w0 + jcol < d) ? 1.0f : val;
      ldsg[row * TILE + jcol] = val;
    }
  }

  __syncthreads();

  // Coalesced b128 writeout: each (g,d) row = 16 floats = 4 float4 = 64 B.
  const float4* lds4 = (const float4*)&tile_out[0][0][0];
  float4* out4 = (float4*)out;
  for (int idx = threadIdx.x; idx < G_ * DMAX * (TILE / 4); idx += 256) {
    const int q  = idx & 3;
    const int d  = (idx >> 2) % DMAX;
    const int gg = idx / (DMAX * (TILE / 4));
    out4[((((size_t)n * G_ + gg) * DMAX + d) * H_ + h) * (W_ / 4) + (w0 >> 2) + q] =
        lds4[(gg * DROWS + d) * (TILE / 4) + q];
  }
}

extern "C" void kernel_launch(void* const* d_in, const int* in_sizes, int n_in,
                              void* d_out, int out_size, void* d_ws, size_t ws_size,
                              hipStream_t stream) {
  (void)in_sizes; (void)n_in; (void)out_size; (void)d_ws; (void)ws_size;
  const float* l = (const float*)d_in[0];
  const float* r = (const float*)d_in[1];
  float* out = (float*)d_out;

  constexpr int EDGE_TILES = DMAX / TILE;          // w0 in {0,16,32}: masking live
  constexpr int FAST_TILES = W_ / TILE - EDGE_TILES;

  dim3 block(256);
  cost_volume_wmma<true ><<<dim3(EDGE_TILES, H_, N_), block, 0, stream>>>(l, r, out, 0);
  cost_volume_wmma<false><<<dim3(FAST_TILES, H_, N_), block, 0, stream>>>(l, r, out, EDGE_TILES);
}